// AdExReservoir_66571993088258
// MI455X (gfx1250) — compile-verified
//
#include <hip/hip_runtime.h>

typedef float v2f __attribute__((ext_vector_type(2)));
typedef float v8f __attribute__((ext_vector_type(8)));

#define N_NEURONS 512
#define N_INPUTS  21
#define BATCH     65536
#define N_STEPS   20
#define DT        0.1f

// One wave32 computes a 16(batch) x 16(neuron) tile:
//   1) I-tile = x[16x21] @ w_in[16x21]^T via 6x V_WMMA_F32_16X16X4_F32
//      (5 full K=4 steps for K=0..19, plus one branch-free tail step for K=20)
//   2) 20-step AdEx dynamics entirely in registers (w-state is dead; only v is output)
//   3) one coalesced store of the 16x16 v-tile
__global__ __launch_bounds__(256) void adex_reservoir_kernel(
    const float* __restrict__ x,        // [BATCH, N_INPUTS]
    const float* __restrict__ alpha,    // [N_NEURONS]
    const float* __restrict__ beta,     // [N_NEURONS]
    const float* __restrict__ delta_t,  // [N_NEURONS]
    const float* __restrict__ w_in,     // [N_NEURONS, N_INPUTS]
    const float* __restrict__ v_thresh, // [N_NEURONS]
    const float* __restrict__ v_reset,  // [N_NEURONS]
    float* __restrict__ out)            // [BATCH, N_NEURONS]
{
    const int lane   = threadIdx.x & 31;
    const int wave   = threadIdx.x >> 5;
    const int tile   = blockIdx.x * 8 + wave;       // 131072 tiles total
    const int tile_n = tile & 31;                   // 32 neuron tiles
    const int tile_m = tile >> 5;                   // 4096 batch tiles
    const int n0     = tile_n * 16;
    const int m0     = tile_m * 16;

    const int half = lane >> 4;                     // 0: lanes 0-15, 1: lanes 16-31
    const int l15  = lane & 15;

    // ---- Input projection tile via f32 WMMA ----
    // A (16x4, f32): lanes [0,16) hold M=lane, K={0,1}; lanes [16,32) hold M=lane-16, K={2,3}
    // B (4x16, f32): lanes [0,16) hold K={0,1} at N=lane; lanes [16,32) hold K={2,3}
    // C/D (16x16, f32): vgpr r holds M=r (lanes 0-15, N=lane) / M=r+8 (lanes 16-31, N=lane-16)
    v8f acc = {};
    const float* __restrict__ xrow = x    + (m0 + l15) * N_INPUTS;  // A row: M = l15
    const float* __restrict__ wrow = w_in + (n0 + l15) * N_INPUTS;  // B col: N = l15

    // 5 full K-blocks: K = 0..19, always in-bounds, unconditional b64 loads
    #pragma unroll
    for (int kk = 0; kk < 5; ++kk) {
        const int kb = kk * 4 + half * 2;           // this lane's K base
        v2f a, b;
        a.x = xrow[kb + 0];
        a.y = xrow[kb + 1];
        b.x = wrow[kb + 0];
        b.y = wrow[kb + 1];
        // (neg_a, A, neg_b, B, c_mod, C, reuse_a, reuse_b)
        acc = __builtin_amdgcn_wmma_f32_16x16x4_f32(false, a, false, b,
                                                    (short)0, acc, false, false);
    }

    // Tail K-block: K = 20..23, only K=20 is real.
    // Lanes with half==0 own K={20,21}: .x = element 20, .y = 0.
    // Lanes with half==1 own K={22,23}: both 0.
    // Unconditional in-bounds load of element 20, then select — no divergent branches.
    {
        const float xa = xrow[20];
        const float wb = wrow[20];
        v2f a, b;
        a.x = (half == 0) ? xa : 0.0f;
        a.y = 0.0f;
        b.x = (half == 0) ? wb : 0.0f;
        b.y = 0.0f;
        acc = __builtin_amdgcn_wmma_f32_16x16x4_f32(false, a, false, b,
                                                    (short)0, acc, false, false);
    }

    // ---- Per-lane neuron parameters (one neuron column per lane) ----
    const int   n      = n0 + l15;
    const float al     = alpha[n];
    const float be     = beta[n];
    const float inv_dt = 1.0f / delta_t[n];
    const float vth    = v_thresh[n];
    const float vrst   = v_reset[n];

    // ---- 20-step AdEx scan, fully register-resident (8 batch rows per lane) ----
    float v[8];
    #pragma unroll
    for (int r = 0; r < 8; ++r) v[r] = 0.0f;

    const float LOG2E = 1.44269504088896340736f;
    for (int t = 0; t < N_STEPS; ++t) {
        #pragma unroll
        for (int r = 0; r < 8; ++r) {
            float ve   = fminf((v[r] - vth) * inv_dt, 10.0f);
            float wexp = be * __builtin_amdgcn_exp2f(ve * LOG2E);   // native v_exp_f32
            float dv   = -al * (v[r] + 70.0f) + acc[r] - wexp;
            float vn   = v[r] + dv * DT;
            v[r] = (vn > vth) ? vrst : vn;                          // spike -> reset
        }
    }

    // ---- Store v tile: vgpr r -> row (m0 + half*8 + r), col n; coalesced per half-wave
    float* __restrict__ o = out + (m0 + half * 8) * N_NEURONS + n;
    #pragma unroll
    for (int r = 0; r < 8; ++r) o[r * N_NEURONS] = v[r];
}

extern "C" void kernel_launch(void* const* d_in, const int* in_sizes, int n_in,
                              void* d_out, int out_size, void* d_ws, size_t ws_size,
                              hipStream_t stream) {
    const float* x        = (const float*)d_in[0];
    const float* alpha    = (const float*)d_in[1];
    const float* beta     = (const float*)d_in[2];
    const float* delta_t  = (const float*)d_in[3];
    const float* w_in     = (const float*)d_in[4];
    const float* v_thresh = (const float*)d_in[5];
    const float* v_reset  = (const float*)d_in[6];
    float* out            = (float*)d_out;

    const int tiles  = (BATCH / 16) * (N_NEURONS / 16);  // 131072
    const int blocks = tiles / 8;                        // 8 waves per 256-thread block

    hipLaunchKernelGGL(adex_reservoir_kernel, dim3(blocks), dim3(256), 0, stream,
                       x, alpha, beta, delta_t, w_in, v_thresh, v_reset, out);
}